// MotionLoss_78864189489380
// MI455X (gfx1250) — compile-verified
//
#include <hip/hip_runtime.h>

#define THREADS 96
#define OWN 94
#define NJ 20  // joints 2..21 stored in LDS

struct LossScales {
  float root3, root6, rot, pos, vel, smooth, contact, floorv, tilt;
};

typedef float v2f __attribute__((ext_vector_type(2)));
typedef float v8f __attribute__((ext_vector_type(8)));

__device__ __forceinline__ void load6(const float* __restrict__ base, int j, float d[6]) {
  const float2* p = reinterpret_cast<const float2*>(base + j * 6);
  float2 x = p[0], y = p[1], z = p[2];
  d[0] = x.x; d[1] = x.y; d[2] = y.x; d[3] = y.y; d[4] = z.x; d[5] = z.y;
}

// 6D continuous rotation -> 3x3 matrix with columns (b1, b2, b3)
__device__ __forceinline__ void cont6d(const float d[6], float R[9]) {
  float n1 = fmaxf(sqrtf(d[0]*d[0] + d[1]*d[1] + d[2]*d[2]), 1e-12f);
  float b1x = d[0]/n1, b1y = d[1]/n1, b1z = d[2]/n1;
  float dt = b1x*d[3] + b1y*d[4] + b1z*d[5];
  float ux = d[3] - dt*b1x, uy = d[4] - dt*b1y, uz = d[5] - dt*b1z;
  float n2 = fmaxf(sqrtf(ux*ux + uy*uy + uz*uz), 1e-12f);
  float b2x = ux/n2, b2y = uy/n2, b2z = uz/n2;
  float b3x = b1y*b2z - b1z*b2y;
  float b3y = b1z*b2x - b1x*b2z;
  float b3z = b1x*b2y - b1y*b2x;
  R[0]=b1x; R[1]=b2x; R[2]=b3x;
  R[3]=b1y; R[4]=b2y; R[5]=b3y;
  R[6]=b1z; R[7]=b2z; R[8]=b3z;
}

// child position = parent_pos + parent_rot @ off ; child rot = parent_rot @ Rj
__device__ __forceinline__ void childState(const float pr[9], const float pp[3],
                                           float ox, float oy, float oz,
                                           const float Rj[9], bool needRot,
                                           float cr[9], float cp[3]) {
  float tx = pp[0] + pr[0]*ox + pr[1]*oy + pr[2]*oz;
  float ty = pp[1] + pr[3]*ox + pr[4]*oy + pr[5]*oz;
  float tz = pp[2] + pr[6]*ox + pr[7]*oy + pr[8]*oz;
  if (needRot) {
    float t[9];
#pragma unroll
    for (int r = 0; r < 3; ++r)
#pragma unroll
      for (int c = 0; c < 3; ++c)
        t[3*r+c] = pr[3*r+0]*Rj[c] + pr[3*r+1]*Rj[3+c] + pr[3*r+2]*Rj[6+c];
#pragma unroll
    for (int k = 0; k < 9; ++k) cr[k] = t[k];
  }
  cp[0] = tx; cp[1] = ty; cp[2] = tz;
}

#define DO_VISIT(J, NEEDROT, PRP, PPP, PRG, PPG, ORP, OPP, ORG, OPG)            \
  do {                                                                          \
    float dP[6], dG[6];                                                         \
    load6(rowP, (J), dP);                                                       \
    load6(rowG, (J), dG);                                                       \
    _Pragma("unroll")                                                           \
    for (int c6 = 0; c6 < 6; ++c6) {                                            \
      float df = dP[c6] - dG[c6];                                               \
      accRot += df * df;                                                        \
    }                                                                           \
    float RP[9], RG[9];                                                         \
    cont6d(dP, RP);                                                             \
    cont6d(dG, RG);                                                             \
    if ((J) == 2) { /* spine tilt: column 1 (b2) of spine matrix */             \
      float t0 = RP[1]-RG[1], t1 = RP[4]-RG[4], t2 = RP[7]-RG[7];               \
      accTilt += t0*t0 + t1*t1 + t2*t2;                                         \
    }                                                                           \
    float ox = off[3*(J)+0], oy = off[3*(J)+1], oz = off[3*(J)+2];              \
    childState(PRP, PPP, ox, oy, oz, RP, (NEEDROT), ORP, OPP);                  \
    childState(PRG, PPG, ox, oy, oz, RG, (NEEDROT), ORG, OPG);                  \
    {                                                                           \
      const int jl = (J) - 2;                                                   \
      _Pragma("unroll")                                                         \
      for (int c3 = 0; c3 < 3; ++c3) {                                          \
        sP[tid][jl][c3] = OPP[c3];                                              \
        sG[tid][jl][c3] = OPG[c3];                                              \
        float df = OPP[c3] - OPG[c3];                                           \
        accPos += df * df;                                                      \
      }                                                                         \
      if ((J) == 18 || (J) == 21) {                                             \
        float df = OPP[1] - OPG[1];                                             \
        accFloor += df * df;                                                    \
      }                                                                         \
    }                                                                           \
  } while (0)

__global__ __launch_bounds__(THREADS) void motion_loss_kernel(
    const float* __restrict__ pred, const float* __restrict__ tgt,
    const float* __restrict__ off, float* __restrict__ out,
    int S, LossScales sc) {
  __shared__ float sP[THREADS][NJ][3];
  __shared__ float sG[THREADS][NJ][3];

  const int tid = threadIdx.x;
  const int b = blockIdx.y;
  const int s = blockIdx.x * OWN + tid;

  float accRoot3 = 0.f, accRoot6 = 0.f, accRot = 0.f, accPos = 0.f;
  float accTilt = 0.f, accFloor = 0.f;
  float accVel = 0.f, accSmooth = 0.f, accContact = 0.f;

  const bool inS = (s < S);
  const bool own = inS && (tid < OWN);

  if (inS) {
    const float* rowP = pred + ((size_t)b * (size_t)S + (size_t)s) * 132u;
    const float* rowG = tgt  + ((size_t)b * (size_t)S + (size_t)s) * 132u;
    __builtin_prefetch(rowP, 0, 3);  // global_prefetch_b8
    __builtin_prefetch(rowG, 0, 3);

    // joint 0: root position channels [0:3)
    {
      float dP[6], dG[6];
      load6(rowP, 0, dP);
      load6(rowG, 0, dG);
#pragma unroll
      for (int c = 0; c < 3; ++c) { float df = dP[c]-dG[c]; accRoot3 += df*df; }
    }
    // joint 1: all 6 channels
    {
      float dP[6], dG[6];
      load6(rowP, 1, dP);
      load6(rowG, 1, dG);
#pragma unroll
      for (int c = 0; c < 6; ++c) { float df = dP[c]-dG[c]; accRoot6 += df*df; }
    }

    const float I9[9] = {1.f,0.f,0.f, 0.f,1.f,0.f, 0.f,0.f,1.f};
    float baseP[3] = { off[3], off[4], off[5] };  // g_pos[1], identical pred/gt

    float cP[9], cPp[3], cG[9], cGp[3];
    float s4P[9], s4Pp[3], s4G[9], s4Gp[3];

    // DFS over the kinematic tree (parent<child ordering, branch point at 4)
    DO_VISIT(2,  true,  I9,  baseP, I9,  baseP, cP, cPp, cG, cGp);
    DO_VISIT(3,  true,  cP,  cPp,  cG,  cGp,  cP, cPp, cG, cGp);
    DO_VISIT(4,  true,  cP,  cPp,  cG,  cGp,  cP, cPp, cG, cGp);
#pragma unroll
    for (int k = 0; k < 9; ++k) { s4P[k] = cP[k]; s4G[k] = cG[k]; }
#pragma unroll
    for (int k = 0; k < 3; ++k) { s4Pp[k] = cPp[k]; s4Gp[k] = cGp[k]; }
    DO_VISIT(5,  true,  cP,  cPp,  cG,  cGp,  cP, cPp, cG, cGp);
    DO_VISIT(6,  false, cP,  cPp,  cG,  cGp,  cP, cPp, cG, cGp);
    DO_VISIT(7,  true,  s4P, s4Pp, s4G, s4Gp, cP, cPp, cG, cGp);
    DO_VISIT(8,  true,  cP,  cPp,  cG,  cGp,  cP, cPp, cG, cGp);
    DO_VISIT(9,  true,  cP,  cPp,  cG,  cGp,  cP, cPp, cG, cGp);
    DO_VISIT(10, false, cP,  cPp,  cG,  cGp,  cP, cPp, cG, cGp);
    DO_VISIT(11, true,  s4P, s4Pp, s4G, s4Gp, cP, cPp, cG, cGp);
    DO_VISIT(12, true,  cP,  cPp,  cG,  cGp,  cP, cPp, cG, cGp);
    DO_VISIT(13, true,  cP,  cPp,  cG,  cGp,  cP, cPp, cG, cGp);
    DO_VISIT(14, false, cP,  cPp,  cG,  cGp,  cP, cPp, cG, cGp);
    DO_VISIT(15, true,  I9,  baseP, I9,  baseP, cP, cPp, cG, cGp);
    DO_VISIT(16, true,  cP,  cPp,  cG,  cGp,  cP, cPp, cG, cGp);
    DO_VISIT(17, true,  cP,  cPp,  cG,  cGp,  cP, cPp, cG, cGp);
    DO_VISIT(18, false, cP,  cPp,  cG,  cGp,  cP, cPp, cG, cGp);
    DO_VISIT(19, true,  I9,  baseP, I9,  baseP, cP, cPp, cG, cGp);
    DO_VISIT(20, true,  cP,  cPp,  cG,  cGp,  cP, cPp, cG, cGp);
    DO_VISIT(21, false, cP,  cPp,  cG,  cGp,  cP, cPp, cG, cGp);
  }

  __syncthreads();

  // halo threads (tid>=OWN) only supply LDS positions; drop their loss terms
  if (!own) {
    accRoot3 = 0.f; accRoot6 = 0.f; accRot = 0.f;
    accPos = 0.f; accTilt = 0.f; accFloor = 0.f;
  }

  const bool v1 = own && (s + 1 < S);
  const bool v2 = own && (s + 2 < S);
  if (v1) {
    for (int j = 0; j < NJ; ++j) {
#pragma unroll
      for (int c = 0; c < 3; ++c) {
        float p0 = sP[tid][j][c],     g0 = sG[tid][j][c];
        float p1 = sP[tid + 1][j][c], g1 = sG[tid + 1][j][c];
        float dv = (p1 - p0) - (g1 - g0);
        accVel += dv * dv;
        if (v2) {
          float p2 = sP[tid + 2][j][c], g2 = sG[tid + 2][j][c];
          float da = (p2 - 2.f*p1 + p0) - (g2 - 2.f*g1 + g0);
          accSmooth += da * da;
        }
      }
    }
    // contact: feet = joints 18, 21 -> LDS rows 16, 19
#pragma unroll
    for (int f = 0; f < 2; ++f) {
      const int j = (f == 0) ? 16 : 19;
      float pvx = sP[tid+1][j][0] - sP[tid][j][0];
      float pvy = sP[tid+1][j][1] - sP[tid][j][1];
      float pvz = sP[tid+1][j][2] - sP[tid][j][2];
      float gvx = sG[tid+1][j][0] - sG[tid][j][0];
      float gvy = sG[tid+1][j][1] - sG[tid][j][1];
      float gvz = sG[tid+1][j][2] - sG[tid][j][2];
      float gn = sqrtf(gvx*gvx + gvy*gvy + gvz*gvz);
      if (gn < 0.005f) accContact += sqrtf(pvx*pvx + pvy*pvy + pvz*pvz);
    }
  }

  float thr = sc.root3*accRoot3 + sc.root6*accRoot6 + sc.rot*accRot
            + sc.pos*accPos + sc.vel*accVel + sc.smooth*accSmooth
            + sc.contact*accContact + sc.floorv*accFloor + sc.tilt*accTilt;

  // Wave-level reduction with V_WMMA_F32_16X16X4_F32:
  // A(16x4) holds lane partials (K=0 for lanes 0-15, K=2 for lanes 16-31),
  // B(4x16) = ones -> D[m][n] = partial_m + partial_{m+16}.
  // Sum of a lane's 8 D-VGPRs gives the half-wave sum on lanes 0 and 16.
  v2f a; a[0] = thr; a[1] = 0.f;
  v2f onesB; onesB[0] = 1.f; onesB[1] = 1.f;
  v8f c = {};
  c = __builtin_amdgcn_wmma_f32_16x16x4_f32(false, a, false, onesB,
                                            (short)0, c, false, false);
  float t = c[0] + c[1] + c[2] + c[3] + c[4] + c[5] + c[6] + c[7];
  const int lane = tid & 31;
  if (lane == 0 || lane == 16) atomicAdd(out, t);
}

__global__ void zero_out_kernel(float* out, int n) {
  for (int i = 0; i < n; ++i) out[i] = 0.f;
}

extern "C" void kernel_launch(void* const* d_in, const int* in_sizes, int n_in,
                              void* d_out, int out_size, void* d_ws, size_t ws_size,
                              hipStream_t stream) {
  (void)n_in; (void)d_ws; (void)ws_size;
  const float* pred = (const float*)d_in[0];
  const float* tgt  = (const float*)d_in[1];
  const float* off  = (const float*)d_in[2];
  float* out = (float*)d_out;

  const int S = 1024, J = 22, C = 6;
  int B = in_sizes[0] / (S * J * C);
  if (B < 1) B = 1;

  const double BS = (double)B * (double)S;
  LossScales sc;
  sc.root3   = (float)(5.0 / (BS * 3.0));
  sc.root6   = (float)(5.0 / (BS * 6.0));
  sc.rot     = (float)(1.0 / (BS * 120.0));
  sc.pos     = (float)(2.0 / (BS * 66.0));
  sc.vel     = (float)(1.0 / ((double)B * (double)(S - 1) * 66.0));
  sc.smooth  = (float)(0.5 / ((double)B * (double)(S - 2) * 66.0));
  sc.contact = (float)(2.0 / ((double)B * (double)(S - 1) * 2.0));
  sc.floorv  = (float)(2.0 / (BS * 2.0));
  sc.tilt    = (float)(1.0 / (BS * 3.0));

  zero_out_kernel<<<1, 1, 0, stream>>>(out, out_size);

  dim3 grid((S + OWN - 1) / OWN, B);
  dim3 block(THREADS);
  motion_loss_kernel<<<grid, block, 0, stream>>>(pred, tgt, off, out, S, sc);
}